// ArcFace_38560216383944
// MI455X (gfx1250) — compile-verified
//
#include <hip/hip_runtime.h>
#include <math.h>

#define S_SCALE 30.0f
#define EPS_F   1e-7f
// cos(0.5), sin(0.5)
#define COS_M   0.8775825618903728f
#define SIN_M   0.4794255386042030f
#define LOG2E   1.4426950408889634f

typedef __attribute__((ext_vector_type(4))) float vf4;
typedef __attribute__((ext_vector_type(2))) float v2f;
typedef __attribute__((ext_vector_type(8))) float v8f;

// One workgroup per row: stream 400KB with NT b128 loads, accumulate
// sum(exp(S*x - S)) = sum(exp2(k*x - k)) per thread, tree-reduce in LDS,
// then thread 0 applies the ArcFace margin correction analytically.
__global__ __launch_bounds__(512)
void arcface_row_kernel(const float* __restrict__ inp,
                        const int* __restrict__ label,
                        float* __restrict__ row_nll,
                        int C) {
  const int row = blockIdx.x;
  const int tid = threadIdx.x;
  const float* __restrict__ rowp = inp + (size_t)row * (size_t)C;
  const int C4 = C >> 2;
  const float k = LOG2E * S_SCALE;

  float s = 0.0f;
  const vf4* __restrict__ rowp4 = (const vf4*)rowp;
  for (int i = tid; i < C4; i += 512) {
    vf4 v = __builtin_nontemporal_load(rowp4 + i);     // global_load_b128 (NT)
    s += exp2f(k * v.x - k);
    s += exp2f(k * v.y - k);
    s += exp2f(k * v.z - k);
    s += exp2f(k * v.w - k);
  }
  // tail (C not multiple of 4) — not hit for C=100000 but keep it general
  for (int i = (C4 << 2) + tid; i < C; i += 512) {
    s += exp2f(k * rowp[i] - k);
  }

  __shared__ float red[512];
  red[tid] = s;
  __syncthreads();
  #pragma unroll
  for (int off = 256; off > 0; off >>= 1) {
    if (tid < off) red[tid] += red[tid + off];
    __syncthreads();
  }

  if (tid == 0) {
    float total = red[0];
    int lab = label[row];
    float t  = rowp[lab];                               // original target cosine
    float tc = fminf(fmaxf(t, -1.0f + EPS_F), 1.0f - EPS_F);
    float sn = sqrtf(fmaxf(0.0f, 1.0f - tc * tc));      // sin(arccos(tc))
    float tl = tc * COS_M - sn * SIN_M;                 // cos(theta + m)
    // swap the target column's exp contribution (shift = S)
    float sum_adj = total - expf(S_SCALE * t - S_SCALE)
                          + expf(S_SCALE * tl - S_SCALE);
    float lse = S_SCALE + logf(sum_adj);
    row_nll[row] = lse - S_SCALE * tl;
  }
}

// Single-wave deterministic reduction of the 512 per-row losses.
// The cross-lane sum is folded through V_WMMA_F32_16X16X4_F32:
// A = diag-striped partials (a1=0), B = ones  =>  D[m,n] = p[m] + p[m+16].
// Lane0 holds D[0..7,0] in d[0..7], lane16 holds D[8..15,0]; one xor-shuffle
// finishes the sum.
__global__ __launch_bounds__(32)
void arcface_finalize(const float* __restrict__ row_nll,
                      float* __restrict__ out, int B) {
  int lane = threadIdx.x;
  float p = 0.0f;
  for (int i = lane; i < B; i += 32) p += row_nll[i];

  v2f a; a.x = p;    a.y = 0.0f;
  v2f b; b.x = 1.0f; b.y = 1.0f;
  v8f c = {};
  v8f d = __builtin_amdgcn_wmma_f32_16x16x4_f32(
      /*neg_a=*/false, a, /*neg_b=*/false, b,
      /*c_mod=*/(short)0, c, /*reuse_a=*/false, /*reuse_b=*/false);

  float acc = d[0] + d[1] + d[2] + d[3] + d[4] + d[5] + d[6] + d[7];
  acc += __shfl_xor(acc, 16, 32);
  if (lane == 0) out[0] = acc / (float)B;
}

extern "C" void kernel_launch(void* const* d_in, const int* in_sizes, int n_in,
                              void* d_out, int out_size, void* d_ws, size_t ws_size,
                              hipStream_t stream) {
  const float* inp  = (const float*)d_in[0];
  const int* label  = (const int*)d_in[1];
  int B = in_sizes[1];                 // 512
  int C = in_sizes[0] / B;             // 100000
  float* row_nll = (float*)d_ws;       // B floats of scratch

  arcface_row_kernel<<<B, 512, 0, stream>>>(inp, label, row_nll, C);
  arcface_finalize<<<1, 32, 0, stream>>>(row_nll, (float*)d_out, B);
}